// ReconDetectionConfidenceMap_39926015984324
// MI455X (gfx1250) — compile-verified
//
#include <hip/hip_runtime.h>
#include <stdint.h>

// Problem shape (fixed by the reference)
#define BNUM 8
#define KNUM 64
#define HNUM 192
#define WNUM 256

// Folded constants:
//   quad  = 0.04*(dy^2+dx^2) - 0.024*dy*dx
//   score = GAUSS_DEN * exp(GAUSS_NUM1 * quad)
//   out   = score / (sum_k score + EPS) = e / (sum_k e + EPS/GAUSS_DEN)
// with e = exp2(A2*(dy^2+dx^2) + B2*dy*dx)      (log2(e) folded -> v_exp_f32)
// Quadratic-in-j form (per row i, per keypoint k):
//   t = C0 + C1*jf + A2*jf^2
//   C1 = B2*dy - 2*A2*cx
//   C0 = A2*(dy^2 + cx^2) - B2*dy*cx
constexpr double kRho = 0.3;
constexpr double kG1  = -1.0 / (2.0 * (1.0 - kRho * kRho));   // GAUSS_NUM1
constexpr double kL2E = 1.4426950408889634;                   // log2(e)
constexpr float  A2f  = (float)(kG1 * 0.04 * kL2E);           // -0.0317075816...
constexpr float  B2f  = (float)(-kG1 * 0.024 * kL2E);         // +0.0190245490...
constexpr float  EPSP = 0.0014984442f;                        // 1e-5 * sqrt((2pi)^2*568.75)

typedef __attribute__((ext_vector_type(4))) unsigned int v4u;
typedef __attribute__((ext_vector_type(8))) int          v8i;
typedef __attribute__((ext_vector_type(4))) int          v4i;

__global__ __launch_bounds__(WNUM)
void recon_confmap_kernel(const float* __restrict__ kp,  // [B,K,2] (x,y)
                          float* __restrict__ out)       // [B,K,H,W]
{
    const int bi = blockIdx.x;            // 0 .. B*H-1 : one block per image row
    const int b  = bi / HNUM;
    const int i  = bi - b * HNUM;
    const int j  = threadIdx.x;           // 0 .. W-1

    __shared__ float2 skp[KNUM];          // raw keypoints for batch b (TDM target)
    __shared__ float2 sC[KNUM];           // per-k quadratic coeffs (C0, C1)

    // ---- Stage keypoints into LDS with the Tensor Data Mover (gfx1250) ----
    // TDM ignores EXEC, so guard to one lane -> exactly one DMA per block.
    if (threadIdx.x == 0) {
        uint64_t gaddr = (uint64_t)(uintptr_t)(kp + (size_t)b * KNUM * 2);
        // flat shared-aperture address: low 32 bits == LDS byte offset
        uint32_t laddr = (uint32_t)(uintptr_t)(&skp[0]);

        v4u g0;                           // D# group 0 (ISA 8.3)
        g0[0] = 1u;                                              // count=1, user desc
        g0[1] = laddr;                                           // lds_addr
        g0[2] = (uint32_t)(gaddr & 0xFFFFFFFFu);                 // global_addr[31:0]
        g0[3] = (uint32_t)((gaddr >> 32) & 0x01FFFFFFu)          // global_addr[56:32]
              | (2u << 30);                                      // type=2 ("image")

        v8i g1;                           // D# group 1 (ISA 8.4)
        g1[0] = (int)(2u << 16);          // wg_mask=0, data_size=2 (4 bytes)
        g1[1] = (int)(128u << 16);        // tensor_dim0 = 128 elements (64 float2)
        g1[2] = (int)(1u << 16);          // tensor_dim1 = 1
        g1[3] = (int)(128u << 16);        // tile_dim0   = 128
        g1[4] = 1;                        // tile_dim1   = 1
        g1[5] = 128;                      // tensor_dim0_stride = 128
        g1[6] = (int)(128u << 16);        // tensor_dim1_stride = 128 (low 16 bits)
        g1[7] = 0;

        v4i g2 = {0, 0, 0, 0};            // unused (<=2D tensor)
        v4i g3 = {0, 0, 0, 0};
        v8i g4 = {0, 0, 0, 0, 0, 0, 0, 0};// extra group (clang-23 6-arg form), unused

        __builtin_amdgcn_tensor_load_to_lds(g0, g1, g2, g3, g4, /*cpol=*/0);
        __builtin_amdgcn_s_wait_tensorcnt(0);   // s_wait_tensorcnt 0
    }
    __syncthreads();

    // ---- Cooperative hoist: per-k quadratic coefficients for this row ----
    if (threadIdx.x < KNUM) {
        const float fi = (float)i;
        float2 c  = skp[threadIdx.x];
        float  cx = c.x;
        float  dy = fi - c.y;
        float  C1 = fmaf(-2.0f * A2f, cx, B2f * dy);
        float  C0 = fmaf(A2f, fmaf(dy, dy, cx * cx), -B2f * dy * cx);
        sC[threadIdx.x] = make_float2(C0, C1);
    }
    __syncthreads();

    const float jf   = (float)j;
    const float base = A2f * jf * jf;     // per-thread constant term

    // ---- Hot loop: 2 VALU + 1 trans + 1 add per score; 64 scores in VGPRs ----
    float e[KNUM];
    float s0 = 0.0f, s1 = 0.0f, s2 = 0.0f, s3 = 0.0f;
#pragma unroll
    for (int k = 0; k < KNUM; k += 4) {
        float2 c0 = sC[k + 0];            // ds_load_b64 broadcast
        float2 c1 = sC[k + 1];
        float2 c2 = sC[k + 2];
        float2 c3 = sC[k + 3];
        float t0 = fmaf(c0.y, jf, c0.x) + base;
        float t1 = fmaf(c1.y, jf, c1.x) + base;
        float t2 = fmaf(c2.y, jf, c2.x) + base;
        float t3 = fmaf(c3.y, jf, c3.x) + base;
        e[k + 0] = __builtin_amdgcn_exp2f(t0);   // v_exp_f32
        e[k + 1] = __builtin_amdgcn_exp2f(t1);
        e[k + 2] = __builtin_amdgcn_exp2f(t2);
        e[k + 3] = __builtin_amdgcn_exp2f(t3);
        s0 += e[k + 0];
        s1 += e[k + 1];
        s2 += e[k + 2];
        s3 += e[k + 3];
    }
    const float sum = (s0 + s1) + (s2 + s3);
    const float inv = 1.0f / (sum + EPSP);

    // ---- Stream 64 coalesced non-temporal stores (128 B/line per wave per k) ----
    float* op = out + (((size_t)b * KNUM) * HNUM + (size_t)i) * WNUM + j;
#pragma unroll
    for (int k = 0; k < KNUM; ++k) {
        __builtin_nontemporal_store(e[k] * inv, op);
        op += (size_t)HNUM * WNUM;
    }
}

extern "C" void kernel_launch(void* const* d_in, const int* in_sizes, int n_in,
                              void* d_out, int out_size, void* d_ws, size_t ws_size,
                              hipStream_t stream) {
    (void)in_sizes; (void)n_in; (void)d_ws; (void)ws_size; (void)out_size;
    const float* keypoints = (const float*)d_in[0];   // [B,K,2] fp32
    float* out = (float*)d_out;                       // [B,K,H,W] fp32

    dim3 grid(BNUM * HNUM);   // 1536 blocks: one per (batch, row)
    dim3 block(WNUM);         // 256 threads = 8 wave32 = one full row
    recon_confmap_kernel<<<grid, block, 0, stream>>>(keypoints, out);
}